// TimeConv_76467597738589
// MI455X (gfx1250) — compile-verified
//
#include <hip/hip_runtime.h>

#define LSLOPE 0.1f

typedef __attribute__((ext_vector_type(2))) float v2f;
typedef __attribute__((ext_vector_type(8))) float v8f;

__device__ __forceinline__ float lrelu(float x) { return x >= 0.f ? x : LSLOPE * x; }

// monotone float<->uint encoding so unsigned atomicMax == float max
__device__ __forceinline__ unsigned fenc(float f) {
    unsigned u = __float_as_uint(f);
    return (u & 0x80000000u) ? ~u : (u | 0x80000000u);
}
__device__ __forceinline__ float fdec(unsigned e) {
    return __uint_as_float((e & 0x80000000u) ? (e ^ 0x80000000u) : ~e);
}

// ---------------------------------------------------------------------------
// K0: init workspace (runs every call; harness poisons ws once, never restores)
// ---------------------------------------------------------------------------
__global__ void k_init(int* po_rank, unsigned* gmax, float* gs, float* gnum,
                       float* msum, float* mdeg, int N, int P_) {
    int i = blockIdx.x * blockDim.x + threadIdx.x;
    if (i < N) po_rank[i] = 0x7FFFFFFF;
    if (i < P_ * 64) { gmax[i] = 0u; gs[i] = 0.f; gnum[i] = 0.f; }
    if (i < P_ * 65) msum[i] = 0.f;
    if (i < P_) mdeg[i] = 0.f;
}

// K1: po_rank[n] = min p with POs[p]==n  (deterministic for duplicate nodes)
__global__ void k_rank(const int* __restrict__ POs, int* __restrict__ po_rank, int P_) {
    int p = blockIdx.x * blockDim.x + threadIdx.x;
    if (p < P_) atomicMin(&po_rank[POs[p]], p);
}

// ---------------------------------------------------------------------------
// K2: h0 = MLP_pi([delay|value]) : [N,4] -> 32 (LeakyReLU) -> 64, via WMMA.
// One wave32 per 16-row tile. GEMM1 is one v_wmma_f32_16x16x4_f32 per 16-col
// tile; layer-1 activations are staged through LDS to convert C-layout ->
// A-layout for GEMM2 (K=32 -> 8 chained WMMAs per 16-col tile, 4 tiles).
// ---------------------------------------------------------------------------
__global__ __launch_bounds__(32) void k_h0_mlp(
    const float* __restrict__ delay, const float* __restrict__ value,
    const float* __restrict__ W1, const float* __restrict__ B1,
    const float* __restrict__ W2, const float* __restrict__ B2,
    float* __restrict__ h0, int N)
{
    __shared__ float Y[16][36];              // 16 rows x 32 cols (+pad)
    const int lane = threadIdx.x;
    const int r    = lane & 15;              // row (A) / col (B,C) within tile
    const int half = lane >> 4;              // k-half selector
    const int tb   = blockIdx.x * 16;        // tile base row
    int row  = tb + r;
    int rowc = row < N ? row : N - 1;

    // A (16x4): lane r holds row r; lanes 0-15 carry K={0,1}, 16-31 K={2,3}.
    // X[row] = [delay0, delay1, value0, value1]
    v2f a1;
    if (half == 0) { a1.x = delay[rowc * 2 + 0]; a1.y = delay[rowc * 2 + 1]; }
    else           { a1.x = value[rowc * 2 + 0]; a1.y = value[rowc * 2 + 1]; }

    // ---- GEMM1: K=4, Nout=32 (two 16-col tiles) ----
    #pragma unroll
    for (int t = 0; t < 2; ++t) {
        int col = t * 16 + r;
        v2f b;
        b.x = W1[(2 * half + 0) * 32 + col];
        b.y = W1[(2 * half + 1) * 32 + col];
        v8f acc = {0.f,0.f,0.f,0.f,0.f,0.f,0.f,0.f};
        acc = __builtin_amdgcn_wmma_f32_16x16x4_f32(false, a1, false, b,
                                                    (short)0, acc, false, false);
        float bias = B1[col];
        #pragma unroll
        for (int v = 0; v < 8; ++v) {
            // C layout: element v in lane -> (M = v + 8*half, Ncol = col)
            Y[v + 8 * half][col] = lrelu(acc[v] + bias);
        }
    }
    __syncthreads();

    // ---- GEMM2: K=32 (8 chained WMMAs), Nout=64 (four 16-col tiles) ----
    #pragma unroll
    for (int nt = 0; nt < 4; ++nt) {
        int col = nt * 16 + r;
        v8f acc = {0.f,0.f,0.f,0.f,0.f,0.f,0.f,0.f};
        #pragma unroll
        for (int ki = 0; ki < 8; ++ki) {
            int k0 = 4 * ki + 2 * half;
            v2f a; a.x = Y[r][k0]; a.y = Y[r][k0 + 1];
            v2f b; b.x = W2[(k0 + 0) * 64 + col]; b.y = W2[(k0 + 1) * 64 + col];
            acc = __builtin_amdgcn_wmma_f32_16x16x4_f32(false, a, false, b,
                                                        (short)0, acc, false, false);
        }
        float bias = B2[col];
        #pragma unroll
        for (int v = 0; v < 8; ++v) {
            int m = tb + v + 8 * half;
            if (m < N) h0[(long)m * 64 + col] = acc[v] + bias;   // no activation
        }
    }
}

// ---------------------------------------------------------------------------
// K3/K4: gate smoothmax aggregation, restricted to dst nodes used by POs.
// ---------------------------------------------------------------------------
__global__ void k_gate_max(const int* __restrict__ src, const int* __restrict__ dst,
                           const int* __restrict__ po_rank,
                           const float* __restrict__ h0,
                           unsigned* __restrict__ gmax, int E, int P_) {
    int e = blockIdx.x * blockDim.x + threadIdx.x;
    if (e >= E) return;
    int slot = po_rank[dst[e]];
    if (slot >= P_) return;
    const float* hs = h0 + (long)src[e] * 64;
    unsigned* g = gmax + (long)slot * 64;
    #pragma unroll 4
    for (int f = 0; f < 64; ++f) atomicMax(&g[f], fenc(hs[f]));
}

__global__ void k_gate_sum(const int* __restrict__ src, const int* __restrict__ dst,
                           const int* __restrict__ po_rank,
                           const float* __restrict__ h0,
                           const unsigned* __restrict__ gmax,
                           float* __restrict__ gs, float* __restrict__ gnum,
                           int E, int P_) {
    int e = blockIdx.x * blockDim.x + threadIdx.x;
    if (e >= E) return;
    int slot = po_rank[dst[e]];
    if (slot >= P_) return;
    const float* hs = h0 + (long)src[e] * 64;
    long base = (long)slot * 64;
    #pragma unroll 4
    for (int f = 0; f < 64; ++f) {
        float v  = hs[f];
        float m  = fdec(gmax[base + f]);
        float ex = expf(v - m);
        atomicAdd(&gs[base + f], ex);
        atomicAdd(&gnum[base + f], ex * v);
    }
}

// ---------------------------------------------------------------------------
// K5: module aggregation. The reference's attention reduces to alpha = 1/deg
// (e - max over width-1 axis == 0), so this is an exact mean of cat(h0,bitpos).
// ---------------------------------------------------------------------------
__global__ void k_mod_sum(const int* __restrict__ src, const int* __restrict__ dst,
                          const int* __restrict__ po_rank,
                          const float* __restrict__ h0,
                          const float* __restrict__ bitpos,
                          float* __restrict__ msum, float* __restrict__ mdeg,
                          int E, int P_) {
    int e = blockIdx.x * blockDim.x + threadIdx.x;
    if (e >= E) return;
    int slot = po_rank[dst[e]];
    if (slot >= P_) return;
    const float* hs = h0 + (long)src[e] * 64;
    long base = (long)slot * 65;
    #pragma unroll 4
    for (int f = 0; f < 64; ++f) atomicAdd(&msum[base + f], hs[f]);
    atomicAdd(&msum[base + 64], bitpos[e]);
    atomicAdd(&mdeg[slot], 1.0f);
}

// ---------------------------------------------------------------------------
// K6: per-PO node MLPs (gate/module), type combine, global MLP, output MLP.
// Only P=1000 rows -> plain VALU, one 64-thread block per PO.
// ---------------------------------------------------------------------------
struct PoW {
    const float *Wg1,*bg1,*Wg2,*bg2;   // neigh_gate   80->32->64
    const float *Wm1,*bm1,*Wm2,*bm2;   // neigh_module 81->32->64
    const float *WG1,*bG1,*WG2,*bG2;   // global        1->32->64
    const float *Wo1,*bo1,*Wo2,*bo2;   // out         128->64->1
};

__global__ __launch_bounds__(64) void k_po_out(
    const float* __restrict__ feat, const float* __restrict__ PO_feat,
    const float* __restrict__ h0,
    const int* __restrict__ POs, const int* __restrict__ is_po,
    const int* __restrict__ is_module, const int* __restrict__ is_pi,
    const int* __restrict__ po_rank,
    const float* __restrict__ gs, const float* __restrict__ gnum,
    const float* __restrict__ msum, const float* __restrict__ mdeg,
    PoW w, float* __restrict__ out, int P_)
{
    int p = blockIdx.x;
    if (p >= P_) return;
    int t = threadIdx.x;
    __shared__ float in_g[80], in_m[81], tg[32], tm[32], t2[32];
    __shared__ float hvec[64], hgv[64], red[64];

    int n = POs[p];
    int slot = po_rank[n];

    if (t < 64) {
        float s = gs[(long)slot * 64 + t];
        in_g[t] = gnum[(long)slot * 64 + t] / (s > 0.f ? s : 1.f);
        float d  = mdeg[slot];
        float dd = d > 0.f ? d : 1.f;
        in_m[t] = msum[(long)slot * 65 + t] / dd;
        if (t == 0) in_m[64] = msum[(long)slot * 65 + 64] / dd;
    }
    if (t < 16) {
        float fv = feat[(long)n * 16 + t];
        in_g[64 + t] = fv;
        in_m[65 + t] = fv;
    }
    __syncthreads();

    if (t < 32) {
        float ag = w.bg1[t], am = w.bm1[t];
        for (int k = 0; k < 80; ++k) ag += in_g[k] * w.Wg1[k * 32 + t];
        for (int k = 0; k < 81; ++k) am += in_m[k] * w.Wm1[k * 32 + t];
        tg[t] = lrelu(ag);
        tm[t] = lrelu(am);
        t2[t] = lrelu(PO_feat[p] * w.WG1[t] + w.bG1[t]);
    }
    __syncthreads();

    {
        float ag = w.bg2[t], am = w.bm2[t], aG = w.bG2[t];
        for (int j = 0; j < 32; ++j) {
            ag += tg[j] * w.Wg2[j * 64 + t];
            am += tm[j] * w.Wm2[j * 64 + t];
            aG += t2[j] * w.WG2[j * 64 + t];
        }
        float hsel = (is_module[n] == 1) ? am : ag;
        if (is_po[n] != 1) hsel = fmaxf(hsel, 0.f);
        float hv = (is_pi[n] == 1) ? h0[(long)n * 64 + t] : hsel;
        hvec[t] = hv;
        hgv[t]  = aG;
    }
    __syncthreads();

    {
        float ay = w.bo1[t];
        for (int k = 0; k < 64; ++k) ay += hvec[k] * w.Wo1[k * 64 + t];
        for (int k = 0; k < 64; ++k) ay += hgv[k] * w.Wo1[(64 + k) * 64 + t];
        red[t] = lrelu(ay) * w.Wo2[t];
    }
    __syncthreads();
    for (int s = 32; s > 0; s >>= 1) {
        if (t < s) red[t] += red[t + s];
        __syncthreads();
    }
    if (t == 0) out[p] = red[0] + w.bo2[0];
}

// ---------------------------------------------------------------------------
// host side
// ---------------------------------------------------------------------------
static int find4(const int* s, int n, int a, int b, int c, int d) {
    for (int i = 0; i + 3 < n; ++i)
        if (s[i] == a && s[i+1] == b && s[i+2] == c && s[i+3] == d) return i;
    return -1;
}

extern "C" void kernel_launch(void* const* d_in, const int* in_sizes, int n_in,
                              void* d_out, int out_size, void* d_ws, size_t ws_size,
                              hipStream_t stream)
{
    // top-level inputs in setup_inputs() dict order
    const float* feat    = (const float*)d_in[0];   // [N,16]
    const float* delay   = (const float*)d_in[1];   // [N,2]
    const float* value   = (const float*)d_in[2];   // [N,2]
    const float* bitpos  = (const float*)d_in[3];   // [EM]
    const float* PO_feat = (const float*)d_in[4];   // [P,1]
    const int* src_g     = (const int*)d_in[n_in - 8];
    const int* dst_g     = (const int*)d_in[n_in - 7];
    const int* src_m     = (const int*)d_in[n_in - 6];
    const int* dst_m     = (const int*)d_in[n_in - 5];
    const int* POs       = (const int*)d_in[n_in - 4];
    const int* is_po     = (const int*)d_in[n_in - 3];
    const int* is_module = (const int*)d_in[n_in - 2];
    const int* is_pi     = (const int*)d_in[n_in - 1];

    int N  = in_sizes[1] / 2;
    int P_ = in_sizes[4];
    int EG = in_sizes[n_in - 8];
    int EM = in_sizes[n_in - 6];

    // locate params by unique (W1,b1,W2,b2) size patterns (order-agnostic)
    int i_pi = find4(in_sizes, n_in, 128, 32, 2048, 64);   if (i_pi < 0) i_pi = 5;
    int i_ng = find4(in_sizes, n_in, 2560, 32, 2048, 64);  if (i_ng < 0) i_ng = 9;
    int i_nm = find4(in_sizes, n_in, 2592, 32, 2048, 64);  if (i_nm < 0) i_nm = 13;
    int i_gl = find4(in_sizes, n_in, 32, 32, 2048, 64);    if (i_gl < 0) i_gl = 26;
    int i_ou = find4(in_sizes, n_in, 8192, 64, 64, 1);     if (i_ou < 0) i_ou = 30;
    // (type / pos / a_m are dead code: softmax over width-1 axis => alpha=1/deg)

    const float* Wp1 = (const float*)d_in[i_pi + 0];
    const float* bp1 = (const float*)d_in[i_pi + 1];
    const float* Wp2 = (const float*)d_in[i_pi + 2];
    const float* bp2 = (const float*)d_in[i_pi + 3];

    PoW w;
    w.Wg1 = (const float*)d_in[i_ng + 0]; w.bg1 = (const float*)d_in[i_ng + 1];
    w.Wg2 = (const float*)d_in[i_ng + 2]; w.bg2 = (const float*)d_in[i_ng + 3];
    w.Wm1 = (const float*)d_in[i_nm + 0]; w.bm1 = (const float*)d_in[i_nm + 1];
    w.Wm2 = (const float*)d_in[i_nm + 2]; w.bm2 = (const float*)d_in[i_nm + 3];
    w.WG1 = (const float*)d_in[i_gl + 0]; w.bG1 = (const float*)d_in[i_gl + 1];
    w.WG2 = (const float*)d_in[i_gl + 2]; w.bG2 = (const float*)d_in[i_gl + 3];
    w.Wo1 = (const float*)d_in[i_ou + 0]; w.bo1 = (const float*)d_in[i_ou + 1];
    w.Wo2 = (const float*)d_in[i_ou + 2]; w.bo2 = (const float*)d_in[i_ou + 3];

    // workspace carve-out
    char* ws = (char*)d_ws;
    size_t off = 0;
    auto carve = [&](size_t bytes) -> void* {
        void* p = ws + off;
        off += (bytes + 255) & ~(size_t)255;
        return p;
    };
    float*    h0      = (float*)    carve((size_t)N * 64 * 4);
    int*      po_rank = (int*)      carve((size_t)N * 4);
    unsigned* gmax    = (unsigned*) carve((size_t)P_ * 64 * 4);
    float*    gs      = (float*)    carve((size_t)P_ * 64 * 4);
    float*    gnum    = (float*)    carve((size_t)P_ * 64 * 4);
    float*    msum    = (float*)    carve((size_t)P_ * 65 * 4);
    float*    mdeg    = (float*)    carve((size_t)P_ * 4);
    (void)ws_size;

    // K0/K1: init + PO rank scatter
    int initN = N > P_ * 65 ? N : P_ * 65;
    k_init<<<(initN + 255) / 256, 256, 0, stream>>>(po_rank, gmax, gs, gnum, msum, mdeg, N, P_);
    k_rank<<<(P_ + 255) / 256, 256, 0, stream>>>(POs, po_rank, P_);

    // K2: h0 via WMMA (one wave per 16-row tile)
    int tiles = (N + 15) / 16;
    k_h0_mlp<<<tiles, 32, 0, stream>>>(delay, value, Wp1, bp1, Wp2, bp2, h0, N);

    // K3/K4: gate smoothmax (filtered to PO dsts)
    k_gate_max<<<(EG + 255) / 256, 256, 0, stream>>>(src_g, dst_g, po_rank, h0, gmax, EG, P_);
    k_gate_sum<<<(EG + 255) / 256, 256, 0, stream>>>(src_g, dst_g, po_rank, h0, gmax, gs, gnum, EG, P_);

    // K5: module mean (filtered to PO dsts)
    k_mod_sum<<<(EM + 255) / 256, 256, 0, stream>>>(src_m, dst_m, po_rank, h0, bitpos, msum, mdeg, EM, P_);

    // K6: per-PO MLPs + combine + output
    k_po_out<<<P_, 64, 0, stream>>>(feat, PO_feat, h0, POs, is_po, is_module, is_pi,
                                    po_rank, gs, gnum, msum, mdeg, w,
                                    (float*)d_out, P_);
    (void)out_size;
}